// SpectralPooling_33071248179379
// MI455X (gfx1250) — compile-verified
//
#include <hip/hip_runtime.h>
#include <math.h>

// CDNA5 (gfx1250) wave32 WMMA fragment types for V_WMMA_F32_16X16X4_F32
typedef float v2f __attribute__((ext_vector_type(2)));
typedef float v8f __attribute__((ext_vector_type(8)));

// LDS intermediate tmp[d'][h][w] strides (floats), chosen for bank-conflict-free
// access: d'-stride % 64 == 8 (so d'+2 shifts banks by 16 for the hi half-lanes
// on stage-2 fragment reads) and h-stride 34 (so h+8 shifts banks by 16 on
// stage-1 accumulator stores).
#define LDS_DSTRIDE 1096
#define LDS_HSTRIDE 34
// total LDS: 64 * 1096 * 4 = 280,576 B  (<= 320 KB per WGP on CDNA5)

// Build A = C32^T * C64[:32,:]  (32 x 64), the composite DCT->truncate->IDCT
// operator applied along the D and H axes. 2048 elements, double-precision
// accumulation for a numerically exact operator.
__global__ void build_A_kernel(float* __restrict__ A) {
    const double PI = 3.14159265358979323846;
    int tid = threadIdx.x; // 256 threads, 8 elements each
    for (int j = 0; j < 8; ++j) {
        int idx = tid + 256 * j;          // idx = d * 64 + dp
        int d  = idx >> 6;
        int dp = idx & 63;
        double acc = 0.0;
        for (int k = 0; k < 32; ++k) {
            double s32 = (k == 0) ? sqrt(1.0 / 32.0) : sqrt(2.0 / 32.0);
            double s64 = (k == 0) ? sqrt(1.0 / 64.0) : sqrt(2.0 / 64.0);
            double c32 = cos(PI * (d  + 0.5) * k / 32.0) * s32; // C32[k][d]
            double c64 = cos(PI * (dp + 0.5) * k / 64.0) * s64; // C64[k][dp]
            acc += c32 * c64;
        }
        A[idx] = (float)acc;
    }
}

// One workgroup per (b,c) slice: out[d,h,w] = sum_{d',h'} A[d,d'] A[h,h'] x[d',h',w]
// Stage 1 (global -> LDS): tmp[d',h,w] = sum_{h'} A[h,h'] x[d',h',w]   (64 GEMMs 32x32x64)
// Stage 2 (LDS -> global): out[d,h,w]  = sum_{d'} A[d,d'] tmp[d',h,w]  (32 GEMMs 32x32x64)
__global__ void __launch_bounds__(256, 1)
spectral_pool_kernel(const float* __restrict__ x, const float* __restrict__ Aop,
                     float* __restrict__ out) {
    __shared__ float tmp[64 * LDS_DSTRIDE];

    const int lane = threadIdx.x & 31;
    const int wave = threadIdx.x >> 5;   // 8 waves
    const int half = lane >> 4;          // hi/lo 16-lane group
    const int l16  = lane & 15;

    const size_t slab = (size_t)blockIdx.x * (64 * 64 * 64); // (b*32+c) slice of x

    // ---- Load the full 32x64 A operator as resident WMMA A-fragments. ----
    // 16x4 f32 A layout: lane(half,m) VGPR v holds A[m][4*kk + v + 2*half].
    v2f aF[2][16];
#pragma unroll
    for (int mt = 0; mt < 2; ++mt) {
#pragma unroll
        for (int kk = 0; kk < 16; ++kk) {
            int row = mt * 16 + l16;
            int col = kk * 4 + 2 * half;
            v2f a;
            a.x = Aop[row * 64 + col];
            a.y = Aop[row * 64 + col + 1];
            aF[mt][kk] = a;
        }
    }

    const v8f vzero = {0.f, 0.f, 0.f, 0.f, 0.f, 0.f, 0.f, 0.f};

    // ============================ Stage 1 ============================
    for (int g = 0; g < 8; ++g) {
        const int dp = g * 8 + wave;                       // d' handled by this wave
        const float* src = x + slab + (size_t)dp * (64 * 64);

        if (g < 7) {
            // Hint the next d'-slab into cache (global_prefetch_b8).
            const float* nxt = x + slab + (size_t)(dp + 8) * (64 * 64);
            __builtin_prefetch(nxt + lane * 128, 0, 1);    // lane -> row 2*lane
        }

        v8f acc00 = vzero, acc01 = vzero, acc10 = vzero, acc11 = vzero;

#pragma unroll
        for (int kk = 0; kk < 16; ++kk) {
            const int r0 = kk * 4 + 2 * half;              // h' row pair base
            // 4x16 f32 B layout: lane(half,n) VGPR v holds B[v + 2*half][n].
            v2f b0, b1;
            b0.x = src[r0 * 64 + l16];
            b0.y = src[(r0 + 1) * 64 + l16];
            b1.x = src[r0 * 64 + 16 + l16];
            b1.y = src[(r0 + 1) * 64 + 16 + l16];

            acc00 = __builtin_amdgcn_wmma_f32_16x16x4_f32(false, aF[0][kk], false, b0,
                                                          (short)0, acc00, false, false);
            acc01 = __builtin_amdgcn_wmma_f32_16x16x4_f32(false, aF[0][kk], false, b1,
                                                          (short)0, acc01, false, false);
            acc10 = __builtin_amdgcn_wmma_f32_16x16x4_f32(false, aF[1][kk], false, b0,
                                                          (short)0, acc10, false, false);
            acc11 = __builtin_amdgcn_wmma_f32_16x16x4_f32(false, aF[1][kk], false, b1,
                                                          (short)0, acc11, false, false);
        }

        // C/D layout: lane(half,n) VGPR v holds C[mt*16 + v + 8*half][nt*16 + n].
        float* trow = &tmp[dp * LDS_DSTRIDE];
#pragma unroll
        for (int v = 0; v < 8; ++v) {
            const int hlo = v + 8 * half;
            trow[hlo * LDS_HSTRIDE + l16]              = acc00[v];
            trow[hlo * LDS_HSTRIDE + 16 + l16]         = acc01[v];
            trow[(16 + hlo) * LDS_HSTRIDE + l16]       = acc10[v];
            trow[(16 + hlo) * LDS_HSTRIDE + 16 + l16]  = acc11[v];
        }
    }

    __syncthreads();

    // ============================ Stage 2 ============================
    for (int g = 0; g < 4; ++g) {
        const int h = g * 8 + wave;                        // h handled by this wave
        const float* tcol = &tmp[h * LDS_HSTRIDE];

        v8f acc00 = vzero, acc01 = vzero, acc10 = vzero, acc11 = vzero;

#pragma unroll
        for (int kk = 0; kk < 16; ++kk) {
            const int r0 = kk * 4 + 2 * half;              // d' row pair base
            v2f b0, b1;
            b0.x = tcol[r0 * LDS_DSTRIDE + l16];
            b0.y = tcol[(r0 + 1) * LDS_DSTRIDE + l16];
            b1.x = tcol[r0 * LDS_DSTRIDE + 16 + l16];
            b1.y = tcol[(r0 + 1) * LDS_DSTRIDE + 16 + l16];

            acc00 = __builtin_amdgcn_wmma_f32_16x16x4_f32(false, aF[0][kk], false, b0,
                                                          (short)0, acc00, false, false);
            acc01 = __builtin_amdgcn_wmma_f32_16x16x4_f32(false, aF[0][kk], false, b1,
                                                          (short)0, acc01, false, false);
            acc10 = __builtin_amdgcn_wmma_f32_16x16x4_f32(false, aF[1][kk], false, b0,
                                                          (short)0, acc10, false, false);
            acc11 = __builtin_amdgcn_wmma_f32_16x16x4_f32(false, aF[1][kk], false, b1,
                                                          (short)0, acc11, false, false);
        }

        // out[b,c,d,h,w] flat: blockIdx.x*32768 + d*1024 + h*32 + w
        float* obase = out + (size_t)blockIdx.x * 32768 + h * 32;
#pragma unroll
        for (int v = 0; v < 8; ++v) {
            const int dlo = v + 8 * half;
            obase[dlo * 1024 + l16]             = acc00[v];
            obase[dlo * 1024 + 16 + l16]        = acc01[v];
            obase[(16 + dlo) * 1024 + l16]      = acc10[v];
            obase[(16 + dlo) * 1024 + 16 + l16] = acc11[v];
        }
    }
}

extern "C" void kernel_launch(void* const* d_in, const int* in_sizes, int n_in,
                              void* d_out, int out_size, void* d_ws, size_t ws_size,
                              hipStream_t stream) {
    const float* x = (const float*)d_in[0];     // (8, 32, 64, 64, 64) f32
    float* A       = (float*)d_ws;              // 32*64 floats = 8 KB scratch
    float* out     = (float*)d_out;             // (8, 32, 32, 32, 32) f32

    build_A_kernel<<<1, 256, 0, stream>>>(A);
    spectral_pool_kernel<<<256, 256, 0, stream>>>(x, A, out);
}